// Loss_Synonymy_19825569038543
// MI455X (gfx1250) — compile-verified
//
#include <hip/hip_runtime.h>
#include <stdint.h>

// ---------------------------------------------------------------------------
// Loss_Synonymy: per-row L2 distance -> tanh -> relu(1 -/+ t) by score -> sum.
// B = 1,048,576 rows, D = 128 f32. Streaming reduction: 1.07 GB in, 4 B out
// => bandwidth-bound (~46 us floor at 23.3 TB/s on MI455X). Data movement uses
// the CDNA5 async global->LDS path (ASYNCcnt) with a 4-deep per-wave pipeline
// and non-temporal hints (each byte is read exactly once).
// ---------------------------------------------------------------------------

#define THREADS 256
#define WAVES_PER_BLOCK (THREADS / 32)
#define NBLK 2048
#define NWAVES (NBLK * WAVES_PER_BLOCK)   // 16384, compile-time stride
#define STAGES 4
#define DDIM 128
#define THRESH 0.6f

typedef float f32x4 __attribute__((ext_vector_type(4)));
typedef __attribute__((address_space(3))) float lds_f32;

__device__ __forceinline__ float wave32_sum(float p) {
    // butterfly reduce across the 32 lanes (wave32 on gfx1250)
    p += __shfl_xor(p, 1, 32);
    p += __shfl_xor(p, 2, 32);
    p += __shfl_xor(p, 4, 32);
    p += __shfl_xor(p, 8, 32);
    p += __shfl_xor(p, 16, 32);
    return p;
}

__global__ __launch_bounds__(THREADS) void syn_loss_partial(
    const float* __restrict__ s1, const float* __restrict__ s2,
    const float* __restrict__ score, float* __restrict__ partial, int B)
{
    // Per-wave 4-deep LDS staging: [wave][stage][matrix][128 floats]
    // 8 waves * 4 stages * 2 matrices * 512 B = 32 KB (of 320 KB/WGP).
    __shared__ float lbuf[WAVES_PER_BLOCK][STAGES][2][DDIM];
    __shared__ float wsum[WAVES_PER_BLOCK];

    const int lane  = threadIdx.x & 31;
    const int wave  = threadIdx.x >> 5;
    const int gwave = blockIdx.x * WAVES_PER_BLOCK + wave;

    // Raw LDS byte offsets (addrspace(3) pointers are 32-bit LDS offsets)
    // for this lane's private 16 B slot in each stage/matrix.
    unsigned ldsoff[STAGES][2];
#pragma unroll
    for (int st = 0; st < STAGES; ++st)
#pragma unroll
        for (int m = 0; m < 2; ++m)
            ldsoff[st][m] =
                (unsigned)(uintptr_t)(lds_f32*)&lbuf[wave][st][m][lane * 4];

    // One wave consumes one row per iteration: 32 lanes * float4 == 128 = D.
    auto issue = [&](int row, int st) {
        const float* g1 = s1 + (size_t)row * DDIM + lane * 4;
        const float* g2 = s2 + (size_t)row * DDIM + lane * 4;
        // WAR guard: make sure prior ds reads of the recycled stage have
        // drained the DS queue (no-op in steady state).
        asm volatile("s_wait_dscnt 0" ::: "memory");
        // CDNA5 async copy global->LDS, 16 B/lane, tracked by ASYNCcnt.
        // NT hint: purely streaming data, touched exactly once.
        asm volatile("global_load_async_to_lds_b128 %0, %1, off th:TH_LOAD_NT"
                     :: "v"(ldsoff[st][0]), "v"(g1) : "memory");
        asm volatile("global_load_async_to_lds_b128 %0, %1, off th:TH_LOAD_NT"
                     :: "v"(ldsoff[st][1]), "v"(g2) : "memory");
    };

    // Iterations this wave runs; NWAVES is constexpr so this is a shift,
    // not a runtime integer division.
    const int niter = (gwave < B) ? ((B - 1 - gwave) / NWAVES + 1) : 0;

    float errAcc = 0.0f;

    // Prologue: fill STAGES-1 pipeline slots.
#pragma unroll
    for (int k = 0; k < STAGES - 1; ++k)
        if (k < niter) issue(gwave + k * NWAVES, k);

    for (int i = 0; i < niter; ++i) {
        const int st  = i & (STAGES - 1);
        const int row = gwave + i * NWAVES;

        if (i + (STAGES - 1) < niter) {
            issue(gwave + (i + STAGES - 1) * NWAVES,
                  (i + STAGES - 1) & (STAGES - 1));
            // 4 iters (8 loads) outstanding; let the newest 3 iters (6 loads)
            // stay in flight -> iteration i's pair is complete.
            asm volatile("s_wait_asynccnt 6" ::: "memory");
        } else {
            const int rem = niter - 1 - i;   // iterations still in flight
            if (rem == 2)      asm volatile("s_wait_asynccnt 4" ::: "memory");
            else if (rem == 1) asm volatile("s_wait_asynccnt 2" ::: "memory");
            else               asm volatile("s_wait_asynccnt 0" ::: "memory");
        }

        // Each lane reads back exactly the 16 B it requested (ds_load_b128).
        f32x4 a = *(const f32x4*)&lbuf[wave][st][0][lane * 4];
        f32x4 b = *(const f32x4*)&lbuf[wave][st][1][lane * 4];
        f32x4 d = a - b;
        float p = d.x * d.x;
        p = __builtin_fmaf(d.y, d.y, p);
        p = __builtin_fmaf(d.z, d.z, p);
        p = __builtin_fmaf(d.w, d.w, p);

        float dist2 = wave32_sum(p);        // full-row sum, all lanes hold it

#if __has_builtin(__builtin_amdgcn_sqrtf)
        float dist = __builtin_amdgcn_sqrtf(dist2);
#else
        float dist = sqrtf(dist2);
#endif
#if __has_builtin(__builtin_amdgcn_tanhf)
        float t = __builtin_amdgcn_tanhf(dist);  // v_tanh_f32 (TRANS) on gfx1250
#else
        float t = tanhf(dist);
#endif
        float sc  = score[row];
        float err = (sc >= THRESH) ? fmaxf(1.0f - t, 0.0f)
                                   : fmaxf(1.0f + t, 0.0f);
        errAcc += err;   // lane-uniform
    }

    if (lane == 0) wsum[wave] = errAcc;
    __syncthreads();
    if (threadIdx.x == 0) {
        float s = 0.0f;
#pragma unroll
        for (int w = 0; w < WAVES_PER_BLOCK; ++w) s += wsum[w];
        partial[blockIdx.x] = s;
    }
}

// Deterministic final reduction: fixed combination order, single block.
__global__ __launch_bounds__(256) void reduce_partials(
    const float* __restrict__ partial, float* __restrict__ out, int n)
{
    __shared__ float sm[256];
    float s = 0.0f;
    for (int i = threadIdx.x; i < n; i += 256) s += partial[i];
    sm[threadIdx.x] = s;
    __syncthreads();
    for (int k = 128; k > 0; k >>= 1) {
        if ((int)threadIdx.x < k) sm[threadIdx.x] += sm[threadIdx.x + k];
        __syncthreads();
    }
    if (threadIdx.x == 0) out[0] = sm[0];
}

extern "C" void kernel_launch(void* const* d_in, const int* in_sizes, int n_in,
                              void* d_out, int out_size, void* d_ws, size_t ws_size,
                              hipStream_t stream) {
    const float* s1    = (const float*)d_in[0];   // [B, 128] f32
    const float* s2    = (const float*)d_in[1];   // [B, 128] f32
    const float* score = (const float*)d_in[2];   // [B] f32
    const int B = in_sizes[2];

    float* partial = (float*)d_ws;                // NBLK floats = 8 KB scratch

    syn_loss_partial<<<NBLK, THREADS, 0, stream>>>(s1, s2, score, partial, B);
    reduce_partials<<<1, 256, 0, stream>>>(partial, (float*)d_out, NBLK);
}